// Pooling_24970939859005
// MI455X (gfx1250) — compile-verified
//
#include <hip/hip_runtime.h>

// Max-pool 2x2 stride 2, VALID, over (N=32, C=64, H=224, W=224) fp32.
// Output (32, 64, 112, 112) fp32.
//
// Pure streaming kernel: every input read once, every output written once.
// Strategy: one thread produces 4 consecutive output pixels of one output row
// ("one quad"). That requires two float4 loads from each of the two source
// input rows (8 input columns) and one float4 store. All accesses are
// 16-byte aligned and fully coalesced. Loads/stores use non-temporal cache
// policy (gfx1250 TH=NT) since the 411MB tensor has zero reuse and exceeds L2.

typedef float v4f __attribute__((ext_vector_type(4)));

constexpr unsigned H  = 224;
constexpr unsigned W  = 224;
constexpr unsigned OH = 112;
constexpr unsigned OW = 112;
constexpr unsigned NC = 32u * 64u;                 // batch * channels images
constexpr unsigned QUADS_PER_ROW = OW / 4;         // 28 float4 stores per output row
constexpr unsigned TOTAL_QUADS = NC * OH * QUADS_PER_ROW;  // 6,422,528 (exact mult of 256)

__global__ __launch_bounds__(256) void
maxpool_2x2_s2_kernel(const float* __restrict__ in, float* __restrict__ out) {
    const unsigned tid = blockIdx.x * blockDim.x + threadIdx.x;
    if (tid >= TOTAL_QUADS) return;   // grid is exact; guard is free

    // Decompose flat quad index. Divisors are compile-time constants so the
    // compiler lowers these to mul-hi + shift (no actual division).
    const unsigned q   = tid % QUADS_PER_ROW;      // which float4 within the output row
    const unsigned row = tid / QUADS_PER_ROW;      // global output row
    const unsigned oy  = row % OH;                 // output row within image
    const unsigned img = row / OH;                 // which (n,c) image

    // Input window: rows 2*oy and 2*oy+1, columns [8q, 8q+8).
    const float* p0 = in + (size_t)img * (H * W) + (size_t)(2u * oy) * W + (size_t)q * 8u;
    const float* p1 = p0 + W;

    // Four 128-bit non-temporal loads (b128, th:NT) — streaming, no L2 rinse.
    const v4f a0 = __builtin_nontemporal_load((const v4f*)p0);
    const v4f a1 = __builtin_nontemporal_load((const v4f*)p0 + 1);
    const v4f b0 = __builtin_nontemporal_load((const v4f*)p1);
    const v4f b1 = __builtin_nontemporal_load((const v4f*)p1 + 1);

    // 2x2 max per output pixel: horizontal pairs within each row, then across rows.
    v4f o;
    o.x = fmaxf(fmaxf(a0.x, a0.y), fmaxf(b0.x, b0.y));
    o.y = fmaxf(fmaxf(a0.z, a0.w), fmaxf(b0.z, b0.w));
    o.z = fmaxf(fmaxf(a1.x, a1.y), fmaxf(b1.x, b1.y));
    o.w = fmaxf(fmaxf(a1.z, a1.w), fmaxf(b1.z, b1.w));

    // One 128-bit non-temporal store of 4 pooled outputs.
    float* po = out + (size_t)img * (OH * OW) + (size_t)oy * OW + (size_t)q * 4u;
    __builtin_nontemporal_store(o, (v4f*)po);
}

extern "C" void kernel_launch(void* const* d_in, const int* in_sizes, int n_in,
                              void* d_out, int out_size, void* d_ws, size_t ws_size,
                              hipStream_t stream) {
    (void)in_sizes; (void)n_in; (void)d_ws; (void)ws_size; (void)out_size;
    const float* in = (const float*)d_in[0];
    float* out = (float*)d_out;

    constexpr unsigned kBlock = 256;                          // 8 wave32 waves
    constexpr unsigned kGrid  = TOTAL_QUADS / kBlock;         // 25,088 blocks, no tail
    maxpool_2x2_s2_kernel<<<kGrid, kBlock, 0, stream>>>(in, out);
}